// MyGCNEdge_49641232007693
// MI455X (gfx1250) — compile-verified
//
#include <hip/hip_runtime.h>
#include <hip/hip_bf16.h>

#define N_NODES 100000
#define N_EDGES 800000
#define F_IN    5
#define F_H1    320
#define F_H2    160
#define F_OUT   80

typedef __attribute__((ext_vector_type(2))) float v2f;
typedef __attribute__((ext_vector_type(8))) float v8f;

// ---------------- degree / normalization ----------------

__global__ void deg_init_kernel(float* deg) {
  int i = blockIdx.x * blockDim.x + threadIdx.x;
  if (i < N_NODES) deg[i] = 1.0f;  // self-loop weight folded in
}

__global__ void deg_scatter_kernel(const int* dst, const float* ew, float* deg) {
  int e = blockIdx.x * blockDim.x + threadIdx.x;
  if (e < N_EDGES) atomicAdd(&deg[dst[e]], ew[e]);
}

__global__ void dinv_kernel(float* deg) {
  int i = blockIdx.x * blockDim.x + threadIdx.x;
  if (i < N_NODES) {
    float d = deg[i];
    deg[i] = (d > 0.0f) ? rsqrtf(d) : 0.0f;   // in-place: deg buffer becomes dinv
  }
}

__global__ void norm_kernel(const int* src, const int* dst, const float* ew,
                            const float* dinv, float* norm) {
  int e = blockIdx.x * blockDim.x + threadIdx.x;
  if (e < N_EDGES) norm[e] = dinv[src[e]] * ew[e] * dinv[dst[e]];
}

// ---------------- conv1 dense part: h1 = x@W1 ; acc1 = b1 + h1*dinv^2 ----------------
// K=5 is too small for WMMA (K step = 4); plain FMA, fully fused with acc1 init.

__global__ void xw1_selfinit_kernel(const float* __restrict__ x,
                                    const float* __restrict__ W1,
                                    const float* __restrict__ b1,
                                    const float* __restrict__ dinv,
                                    float* __restrict__ h1,
                                    float* __restrict__ acc1) {
  int t = blockIdx.x * blockDim.x + threadIdx.x;
  if (t >= N_NODES * F_H1) return;
  int n = t / F_H1;
  int f = t - n * F_H1;           // f fastest-varying -> coalesced stores
  const float* xr = x + n * F_IN;
  float s = 0.0f;
#pragma unroll
  for (int k = 0; k < F_IN; ++k) s = fmaf(xr[k], W1[k * F_H1 + f], s);
  h1[t] = s;
  float di = dinv[n];
  acc1[t] = fmaf(s, di * di, b1[f]);   // bias + self-loop contribution
}

// ---------------- edge scatter: acc[dst] += h[src] * norm, one wave per edge ----------------

template <int F>
__global__ void edge_scatter_kernel(const int* __restrict__ src,
                                    const int* __restrict__ dst,
                                    const float* __restrict__ norm,
                                    const float* __restrict__ h,
                                    float* __restrict__ acc) {
  int wave = (blockIdx.x * blockDim.x + threadIdx.x) >> 5;
  int lane = threadIdx.x & 31;
  if (wave >= N_EDGES) return;
  int s = src[wave];
  int d = dst[wave];
  float w = norm[wave];
  const float* hs = h + (size_t)s * F;
  float* ad = acc + (size_t)d * F;
#pragma unroll
  for (int f = lane; f < F; f += 32)
    atomicAdd(&ad[f], hs[f] * w);
}

// ---------------- WMMA f32 GEMM, wave-per-16x16-tile ----------------
// A fragment (16x4 f32): lane L: m = m0 + (L%16); a = {A[m][k0+2h], A[m][k0+2h+1]}, h = L/16
// B fragment (4x16 f32): lane L: n = n0 + (L%16); b = {B[k0+2h][n], B[k0+2h+1][n]}
// C/D (16x16 f32, 8 VGPR): VGPR v, lane L -> D[m0 + v + 8h][n0 + (L%16)]

// conv2 GEMM: h2 = relu(acc1) @ W2 ; fused epilogue writes h2 and acc2 = b2 + h2*dinv^2
__global__ void __launch_bounds__(256)
gemm2_wmma_kernel(const float* __restrict__ A,   // acc1 [N, 320]
                  const float* __restrict__ B,   // W2   [320, 160]
                  const float* __restrict__ b2,
                  const float* __restrict__ dinv,
                  float* __restrict__ h2,        // [N, 160]
                  float* __restrict__ acc2) {    // [N, 160]
  const int NT = F_H2 / 16;                       // 10 N-tiles
  const int numTiles = (N_NODES / 16) * NT;       // 62500 waves
  int wave = (blockIdx.x * blockDim.x + threadIdx.x) >> 5;
  if (wave >= numTiles) return;                   // wave-uniform: EXEC stays all-1s
  int lane = threadIdx.x & 31;
  int r = lane & 15;
  int half = lane >> 4;
  int tm = wave / NT;
  int tn = wave - tm * NT;
  int m0 = tm * 16, n0 = tn * 16;

  const float* arow = A + (size_t)(m0 + r) * F_H1;
  const float* bcol = B + (n0 + r);
  v8f c = {};
  for (int k0 = 0; k0 < F_H1; k0 += 4) {
    int ka = k0 + 2 * half;
    v2f a, b;
    a.x = fmaxf(arow[ka], 0.0f);                  // fused relu on conv1 output
    a.y = fmaxf(arow[ka + 1], 0.0f);
    b.x = bcol[(size_t)ka * F_H2];
    b.y = bcol[(size_t)(ka + 1) * F_H2];
    c = __builtin_amdgcn_wmma_f32_16x16x4_f32(false, a, false, b, (short)0, c,
                                              false, false);
  }
  int f = n0 + r;
  float bb = b2[f];
#pragma unroll
  for (int v = 0; v < 8; ++v) {
    int m = m0 + v + 8 * half;
    float d = c[v];
    h2[(size_t)m * F_H2 + f] = d;
    float di = dinv[m];
    acc2[(size_t)m * F_H2 + f] = fmaf(d, di * di, bb);  // bias + self-loop
  }
}

// embed GEMM: out = acc2 @ We + be
__global__ void __launch_bounds__(256)
gemm3_wmma_kernel(const float* __restrict__ A,   // acc2 [N, 160]
                  const float* __restrict__ B,   // We   [160, 80]
                  const float* __restrict__ be,
                  float* __restrict__ out) {     // [N, 80]
  const int NT = F_OUT / 16;                      // 5 N-tiles
  const int numTiles = (N_NODES / 16) * NT;       // 31250 waves
  int wave = (blockIdx.x * blockDim.x + threadIdx.x) >> 5;
  if (wave >= numTiles) return;
  int lane = threadIdx.x & 31;
  int r = lane & 15;
  int half = lane >> 4;
  int tm = wave / NT;
  int tn = wave - tm * NT;
  int m0 = tm * 16, n0 = tn * 16;

  const float* arow = A + (size_t)(m0 + r) * F_H2;
  const float* bcol = B + (n0 + r);
  v8f c = {};
  for (int k0 = 0; k0 < F_H2; k0 += 4) {
    int ka = k0 + 2 * half;
    v2f a, b;
    a.x = arow[ka];
    a.y = arow[ka + 1];
    b.x = bcol[(size_t)ka * F_OUT];
    b.y = bcol[(size_t)(ka + 1) * F_OUT];
    c = __builtin_amdgcn_wmma_f32_16x16x4_f32(false, a, false, b, (short)0, c,
                                              false, false);
  }
  int f = n0 + r;
  float bb = be[f];
#pragma unroll
  for (int v = 0; v < 8; ++v) {
    int m = m0 + v + 8 * half;
    out[(size_t)m * F_OUT + f] = c[v] + bb;
  }
}

// ---------------- launch ----------------

extern "C" void kernel_launch(void* const* d_in, const int* in_sizes, int n_in,
                              void* d_out, int out_size, void* d_ws, size_t ws_size,
                              hipStream_t stream) {
  (void)in_sizes; (void)n_in; (void)out_size; (void)ws_size;
  const float* x  = (const float*)d_in[0];
  const int*   ei = (const int*)d_in[1];
  const float* ew = (const float*)d_in[2];
  const float* W1 = (const float*)d_in[3];
  const float* b1 = (const float*)d_in[4];
  const float* W2 = (const float*)d_in[5];
  const float* b2 = (const float*)d_in[6];
  const float* We = (const float*)d_in[7];
  const float* be = (const float*)d_in[8];
  float* out = (float*)d_out;
  const int* src = ei;
  const int* dst = ei + N_EDGES;

  // workspace layout (256B aligned):
  //   dinv : N floats        (400,000 B -> pad 400,128)
  //   norm : E floats        (3,200,000 B)
  //   bufA : N*320 floats    (128,000,000 B)  h1, later h2(64MB) + acc2(64MB)
  //   bufB : N*320 floats    (128,000,000 B)  acc1
  char* ws = (char*)d_ws;
  float* dinv = (float*)(ws);
  float* norm = (float*)(ws + 400128);
  float* bufA = (float*)(ws + 400128 + 3200000);
  float* bufB = (float*)(ws + 400128 + 3200000 + 128000000);
  float* h1   = bufA;
  float* acc1 = bufB;
  float* h2   = bufA;                                 // reuse (h1 dead after scatter1)
  float* acc2 = bufA + (size_t)N_NODES * F_H2;        // second half of bufA

  const int T = 256;
  deg_init_kernel<<<(N_NODES + T - 1) / T, T, 0, stream>>>(dinv);
  deg_scatter_kernel<<<(N_EDGES + T - 1) / T, T, 0, stream>>>(dst, ew, dinv);
  dinv_kernel<<<(N_NODES + T - 1) / T, T, 0, stream>>>(dinv);
  norm_kernel<<<(N_EDGES + T - 1) / T, T, 0, stream>>>(src, dst, ew, dinv, norm);

  // conv1 dense + fused bias/self-loop init
  xw1_selfinit_kernel<<<(N_NODES * F_H1 + T - 1) / T, T, 0, stream>>>(
      x, W1, b1, dinv, h1, acc1);
  // conv1 edge scatter (wave per edge)
  edge_scatter_kernel<F_H1><<<(N_EDGES * 32 + T - 1) / T, T, 0, stream>>>(
      src, dst, norm, h1, acc1);

  // conv2: WMMA GEMM with fused relu + bias/self-loop epilogue
  {
    long long waves = (long long)(N_NODES / 16) * (F_H2 / 16);
    gemm2_wmma_kernel<<<(int)((waves * 32 + T - 1) / T), T, 0, stream>>>(
        acc1, W2, b2, dinv, h2, acc2);
  }
  // conv2 edge scatter
  edge_scatter_kernel<F_H2><<<(N_EDGES * 32 + T - 1) / T, T, 0, stream>>>(
      src, dst, norm, h2, acc2);

  // embed linear: WMMA GEMM + bias
  {
    long long waves = (long long)(N_NODES / 16) * (F_OUT / 16);
    gemm3_wmma_kernel<<<(int)((waves * 32 + T - 1) / T), T, 0, stream>>>(
        acc2, We, be, out);
  }
}